// GATGRULayer_43499428774615
// MI455X (gfx1250) — compile-verified
//
#include <hip/hip_runtime.h>
#include <hip/hip_bf16.h>
#include <math.h>

// ---------------------------------------------------------------------------
// GATGRULayer fused implementation for gfx1250 (MI455X), wave32 + WMMA f32.
//   Phase 1: fused 2-layer GRU over T=12, batch 16384, feat 32 -> hid 64.
//   Phase 2: flash-style GAT aggregation per (layer, bc) pair.
// ---------------------------------------------------------------------------

typedef __attribute__((ext_vector_type(2))) float v2f;
typedef __attribute__((ext_vector_type(8))) float v8f;

#define T_SEQ   12
#define BCN     16
#define NODES   1024
#define FEAT    32
#define HID     64
#define G3      192            // 3*HID, PyTorch gate order r,z,n
#define NROWS   (BCN * NODES)  // 16384
#define NEG_BIG (-1e16f)

// D = A(16x4 f32) * B(4x16 f32) + C(16x16 f32)
__device__ __forceinline__ v8f wmma_f32(v2f a, v2f b, v8f c) {
  return __builtin_amdgcn_wmma_f32_16x16x4_f32(
      /*neg_a=*/false, a, /*neg_b=*/false, b,
      /*c_mod=*/(short)0, c, /*reuse_a=*/false, /*reuse_b=*/false);
}

__device__ __forceinline__ v8f bias8(float v) {
  v8f r;
#pragma unroll
  for (int i = 0; i < 8; ++i) r[i] = v;
  return r;
}

__device__ __forceinline__ float sigmoidf_(float x) {
  return 1.0f / (1.0f + __expf(-x));
}

// ---------------------------------------------------------------------------
// Fused 2-layer GRU. 256 blocks x 128 threads (4 waves); each wave owns 16
// rows for all 12 timesteps. Weights live transposed in LDS (CDNA5 320KB/WGP).
// h-states round-trip through padded LDS slabs to convert D-layout -> A-layout
// between the elementwise gate update and the next step's h @ W_hh^T.
// ---------------------------------------------------------------------------
#define GRU_WAVES 4
#define HPAD 66   // padded row stride (dwords): bank = (2*row + col) mod 64

__global__ __launch_bounds__(128) void gru_fused_kernel(
    const float* __restrict__ x_in,   // [12, 16384, 32]
    const float* __restrict__ w_ih0,  // [192, 32]
    const float* __restrict__ w_hh0,  // [192, 64]
    const float* __restrict__ b_ih0,  // [192]
    const float* __restrict__ b_hh0,  // [192]
    const float* __restrict__ w_ih1,  // [192, 64]
    const float* __restrict__ w_hh1,  // [192, 64]
    const float* __restrict__ b_ih1,  // [192]
    const float* __restrict__ b_hh1,  // [192]
    float* __restrict__ z_out)        // [2, 16384, 64]
{
  __shared__ float wih0_t[FEAT * G3];               // [k][g], 24 KB
  __shared__ float whh0_t[HID * G3];                // 48 KB
  __shared__ float wih1_t[HID * G3];                // 48 KB
  __shared__ float whh1_t[HID * G3];                // 48 KB
  __shared__ float hst[2][GRU_WAVES][16 * HPAD];    // h staging, ~33 KB

  const int tid = threadIdx.x;
  // Cooperative transposed weight staging: Wt[k][g] = W[g][k]
  for (int i = tid; i < FEAT * G3; i += blockDim.x) {
    const int k = i / G3, g = i % G3;
    wih0_t[i] = w_ih0[g * FEAT + k];
  }
  for (int i = tid; i < HID * G3; i += blockDim.x) {
    const int k = i / G3, g = i % G3;
    whh0_t[i] = w_hh0[g * HID + k];
    wih1_t[i] = w_ih1[g * HID + k];
    whh1_t[i] = w_hh1[g * HID + k];
  }

  const int wave = tid >> 5;
  const int lane = tid & 31;
  const int l16  = lane & 15;
  const int lh   = lane >> 4;
  float* h0s = &hst[0][wave][0];
  float* h1s = &hst[1][wave][0];
  for (int i = lane; i < 16 * HPAD; i += 32) { h0s[i] = 0.0f; h1s[i] = 0.0f; }
  __syncthreads();

  const int row0 = blockIdx.x * (GRU_WAVES * 16) + wave * 16;

  for (int t = 0; t < T_SEQ; ++t) {
    // ---- A-matrix fragments (16-row tile), ISA f32 16x4 layout ----
    v2f xf[8];
    {
      const float* xrow = x_in + ((size_t)t * NROWS + row0 + l16) * FEAT;
#pragma unroll
      for (int ks = 0; ks < 8; ++ks) {
        const int k = 4 * ks + 2 * lh;
        xf[ks].x = xrow[k]; xf[ks].y = xrow[k + 1];
      }
    }
    v2f h0f[16], h1f[16];
#pragma unroll
    for (int ks = 0; ks < 16; ++ks) {
      const int k = 4 * ks + 2 * lh;
      h0f[ks].x = h0s[l16 * HPAD + k]; h0f[ks].y = h0s[l16 * HPAD + k + 1];
      h1f[ks].x = h1s[l16 * HPAD + k]; h1f[ks].y = h1s[l16 * HPAD + k + 1];
    }

    // ================= layer 0 =================
#pragma unroll
    for (int j = 0; j < 4; ++j) {
      const int gr = j * 16 + l16;            // r-gate column
      const int gz = 64 + j * 16 + l16;       // z-gate column
      const int gn = 128 + j * 16 + l16;      // n-gate column
      v8f gir = bias8(b_ih0[gr]), giz = bias8(b_ih0[gz]), gin = bias8(b_ih0[gn]);
      v8f ghr = bias8(b_hh0[gr]), ghz = bias8(b_hh0[gz]), ghn = bias8(b_hh0[gn]);
#pragma unroll
      for (int ks = 0; ks < 8; ++ks) {        // gi = x @ W_ih0^T, K = 32
        const int k = 4 * ks + 2 * lh;
        v2f br, bz, bn;
        br.x = wih0_t[k * G3 + gr]; br.y = wih0_t[(k + 1) * G3 + gr];
        bz.x = wih0_t[k * G3 + gz]; bz.y = wih0_t[(k + 1) * G3 + gz];
        bn.x = wih0_t[k * G3 + gn]; bn.y = wih0_t[(k + 1) * G3 + gn];
        gir = wmma_f32(xf[ks], br, gir);
        giz = wmma_f32(xf[ks], bz, giz);
        gin = wmma_f32(xf[ks], bn, gin);
      }
#pragma unroll
      for (int ks = 0; ks < 16; ++ks) {       // gh = h0 @ W_hh0^T, K = 64
        const int k = 4 * ks + 2 * lh;
        v2f br, bz, bn;
        br.x = whh0_t[k * G3 + gr]; br.y = whh0_t[(k + 1) * G3 + gr];
        bz.x = whh0_t[k * G3 + gz]; bz.y = whh0_t[(k + 1) * G3 + gz];
        bn.x = whh0_t[k * G3 + gn]; bn.y = whh0_t[(k + 1) * G3 + gn];
        ghr = wmma_f32(h0f[ks], br, ghr);
        ghz = wmma_f32(h0f[ks], bz, ghz);
        ghn = wmma_f32(h0f[ks], bn, ghn);
      }
      // gate math in D-layout: row = v + 8*lh, col = 16j + l16
#pragma unroll
      for (int v = 0; v < 8; ++v) {
        const int a = (v + 8 * lh) * HPAD + j * 16 + l16;
        const float hold = h0s[a];
        const float r  = sigmoidf_(gir[v] + ghr[v]);
        const float zg = sigmoidf_(giz[v] + ghz[v]);
        const float n  = tanhf(gin[v] + r * ghn[v]);
        h0s[a] = (1.0f - zg) * n + zg * hold;
      }
    }

    // Reload layer-0 output as A-fragments (layer-1 input)
#pragma unroll
    for (int ks = 0; ks < 16; ++ks) {
      const int k = 4 * ks + 2 * lh;
      h0f[ks].x = h0s[l16 * HPAD + k]; h0f[ks].y = h0s[l16 * HPAD + k + 1];
    }

    // ================= layer 1 =================
#pragma unroll
    for (int j = 0; j < 4; ++j) {
      const int gr = j * 16 + l16;
      const int gz = 64 + j * 16 + l16;
      const int gn = 128 + j * 16 + l16;
      v8f gir = bias8(b_ih1[gr]), giz = bias8(b_ih1[gz]), gin = bias8(b_ih1[gn]);
      v8f ghr = bias8(b_hh1[gr]), ghz = bias8(b_hh1[gz]), ghn = bias8(b_hh1[gn]);
#pragma unroll
      for (int ks = 0; ks < 16; ++ks) {       // gi = h0_new @ W_ih1^T, K = 64
        const int k = 4 * ks + 2 * lh;
        v2f br, bz, bn;
        br.x = wih1_t[k * G3 + gr]; br.y = wih1_t[(k + 1) * G3 + gr];
        bz.x = wih1_t[k * G3 + gz]; bz.y = wih1_t[(k + 1) * G3 + gz];
        bn.x = wih1_t[k * G3 + gn]; bn.y = wih1_t[(k + 1) * G3 + gn];
        gir = wmma_f32(h0f[ks], br, gir);
        giz = wmma_f32(h0f[ks], bz, giz);
        gin = wmma_f32(h0f[ks], bn, gin);
      }
#pragma unroll
      for (int ks = 0; ks < 16; ++ks) {       // gh = h1 @ W_hh1^T, K = 64
        const int k = 4 * ks + 2 * lh;
        v2f br, bz, bn;
        br.x = whh1_t[k * G3 + gr]; br.y = whh1_t[(k + 1) * G3 + gr];
        bz.x = whh1_t[k * G3 + gz]; bz.y = whh1_t[(k + 1) * G3 + gz];
        bn.x = whh1_t[k * G3 + gn]; bn.y = whh1_t[(k + 1) * G3 + gn];
        ghr = wmma_f32(h1f[ks], br, ghr);
        ghz = wmma_f32(h1f[ks], bz, ghz);
        ghn = wmma_f32(h1f[ks], bn, ghn);
      }
#pragma unroll
      for (int v = 0; v < 8; ++v) {
        const int a = (v + 8 * lh) * HPAD + j * 16 + l16;
        const float hold = h1s[a];
        const float r  = sigmoidf_(gir[v] + ghr[v]);
        const float zg = sigmoidf_(giz[v] + ghz[v]);
        const float n  = tanhf(gin[v] + r * ghn[v]);
        h1s[a] = (1.0f - zg) * n + zg * hold;
      }
    }
  }

  // z[0] = hT0, z[1] = hT1  ([2, 16384, 64])
  for (int i = lane; i < 16 * HID; i += 32) {
    const int r = i / HID, c = i % HID;
    z_out[(size_t)(row0 + r) * HID + c] = h0s[r * HPAD + c];
    z_out[(size_t)NROWS * HID + (size_t)(row0 + r) * HID + c] = h1s[r * HPAD + c];
  }
}

// ---------------------------------------------------------------------------
// Flash-style GAT aggregation. One wave per (pair, 16-row tile).
// Scores stored transposed in LDS as S[m][row] (64 KB) so both the softmax
// sweep and the WMMA-A reads for P@z are lane-contiguous; the 1024x1024
// attention matrix never touches HBM.
// ---------------------------------------------------------------------------
__global__ __launch_bounds__(32) void gat_kernel(
    const float* __restrict__ z,     // [2, 16384, 64] (= [pair][1024][64])
    const int*   __restrict__ mask,  // [1024, 1024]
    float*       __restrict__ out)   // [2, 16, 1024, 64]
{
  __shared__ float S[NODES * 16];    // S[m][row], 64 KB

  const int lane  = threadIdx.x;
  const int l16   = lane & 15;
  const int lh    = lane >> 4;
  const int ntile = blockIdx.x & 63;
  const int pair  = blockIdx.x >> 6;          // l*16 + b
  const float* zp = z + (size_t)pair * NODES * HID;
  const int n0 = ntile * 16;

  // A fragments: rows n0..n0+15 of z, all 64 K-columns (held for whole kernel)
  v2f za[16];
  {
    const float* zrow = zp + (size_t)(n0 + l16) * HID;
#pragma unroll
    for (int ks = 0; ks < 16; ++ks) {
      const int k = 4 * ks + 2 * lh;
      za[ks].x = zrow[k]; za[ks].y = zrow[k + 1];
    }
  }

  // ---- Phase A: scores S[n, m] = masked leaky_relu(z[n] . z[m]) ----
  for (int mt = 0; mt < 64; ++mt) {
    v8f acc = {};
#pragma unroll
    for (int ks = 0; ks < 16; ++ks) {
      const int k = 4 * ks + 2 * lh;
      v2f b;                                   // B[k][m] = z[m][k]
      b.x = zp[(size_t)(mt * 16 + l16) * HID + k];
      b.y = zp[(size_t)(mt * 16 + l16) * HID + k + 1];
      acc = wmma_f32(za[ks], b, acc);
    }
    const int col = mt * 16 + l16;
#pragma unroll
    for (int v = 0; v < 8; ++v) {
      const int row = v + 8 * lh;
      const float s  = acc[v];
      const float lr = s > 0.0f ? s : 0.1f * s;
      float a = lr * (float)mask[(size_t)(n0 + row) * NODES + col];
      a = (a == 0.0f) ? NEG_BIG : a;           // masked_fill(att==0, -1e16)
      S[col * 16 + row] = a;
    }
  }

  // ---- Phase B: row softmax; lane pair (r, r+16) splits the 1024 columns ----
  {
    const int r = l16;
    const int mbase = lh * 512;
    float mx = -INFINITY;
    for (int i = 0; i < 512; ++i) mx = fmaxf(mx, S[(mbase + i) * 16 + r]);
    mx = fmaxf(mx, __shfl_xor(mx, 16, 32));
    float sum = 0.0f;
    for (int i = 0; i < 512; ++i) sum += __expf(S[(mbase + i) * 16 + r] - mx);
    sum += __shfl_xor(sum, 16, 32);
    const float inv = 1.0f / sum;
    for (int i = 0; i < 512; ++i) {
      const int idx = (mbase + i) * 16 + r;
      S[idx] = __expf(S[idx] - mx) * inv;
    }
  }

  // ---- Phase C: out = P @ z  (A = P from LDS, B = z from L2) ----
  v8f accO[4] = {};
  for (int mt = 0; mt < 64; ++mt) {
#pragma unroll
    for (int ks = 0; ks < 4; ++ks) {
      const int k = mt * 16 + 4 * ks + 2 * lh;
      v2f a;                                   // A[row][k] = P[row][k]
      a.x = S[k * 16 + l16];
      a.y = S[(k + 1) * 16 + l16];
#pragma unroll
      for (int ct = 0; ct < 4; ++ct) {
        v2f b;                                 // B[k][h] = z[k][h]
        b.x = zp[(size_t)k * HID + ct * 16 + l16];
        b.y = zp[(size_t)(k + 1) * HID + ct * 16 + l16];
        accO[ct] = wmma_f32(a, b, accO[ct]);
      }
    }
  }

  float* op = out + (size_t)pair * NODES * HID;
#pragma unroll
  for (int ct = 0; ct < 4; ++ct)
#pragma unroll
    for (int v = 0; v < 8; ++v) {
      const int row = v + 8 * lh;
      op[(size_t)(n0 + row) * HID + ct * 16 + l16] = accO[ct][v];
    }
}

// ---------------------------------------------------------------------------
extern "C" void kernel_launch(void* const* d_in, const int* in_sizes, int n_in,
                              void* d_out, int out_size, void* d_ws, size_t ws_size,
                              hipStream_t stream) {
  const float* x     = (const float*)d_in[0];   // h: [12,16,1024,32]
  const int*   mask  = (const int*)  d_in[1];   // [1024,1024]
  const float* w_ih0 = (const float*)d_in[2];
  const float* w_hh0 = (const float*)d_in[3];
  const float* b_ih0 = (const float*)d_in[4];
  const float* b_hh0 = (const float*)d_in[5];
  const float* w_ih1 = (const float*)d_in[6];
  const float* w_hh1 = (const float*)d_in[7];
  const float* b_ih1 = (const float*)d_in[8];
  const float* b_hh1 = (const float*)d_in[9];
  float* out  = (float*)d_out;                  // [2,16,1024,64]
  float* zbuf = (float*)d_ws;                   // [2,16384,64] = 8 MB scratch

  // 256 blocks x 4 waves x 16 rows = 16384 rows, exact.
  gru_fused_kernel<<<256, 128, 0, stream>>>(x, w_ih0, w_hh0, b_ih0, b_hh0,
                                            w_ih1, w_hh1, b_ih1, b_hh1, zbuf);
  // 32 pairs x 64 row-tiles, one wave each.
  gat_kernel<<<2048, 32, 0, stream>>>(zbuf, mask, out);
}